// UFGConv_69724499083641
// MI455X (gfx1250) — compile-verified
//
#include <hip/hip_runtime.h>
#include <hip/hip_bf16.h>

typedef float v2f __attribute__((ext_vector_type(2)));
typedef float v8f __attribute__((ext_vector_type(8)));

#define NNODES   50000
#define FEAT     256
#define NNZ_OP   800000
#define THRESH   1e-4f

// ---------------------------------------------------------------------------
// GEMM: C[N,256] = A[N,256] * B[256,256] using fp32 WMMA 16x16x4.
// One wave computes one 16x16 tile of C. Block = 256 threads = 8 waves,
// grid = (N/16, 2): wave w in block (bx, by) covers rows bx*16..+16,
// cols ((by*8+w)*16)..+16.
// ---------------------------------------------------------------------------
__global__ __launch_bounds__(256)
void gemm_f32_wmma(const float* __restrict__ A,
                   const float* __restrict__ B,
                   float* __restrict__ C) {
    const int wave = threadIdx.x >> 5;
    const int lane = threadIdx.x & 31;
    const int m0   = blockIdx.x * 16;
    const int n0   = (blockIdx.y * 8 + wave) * 16;
    const int mr   = lane & 15;     // row within tile (A) / col within tile (B,C)
    const int kh   = lane >> 4;     // K-half selector (0 or 1)

    const float* Arow = A + (size_t)(m0 + mr) * FEAT;

    v8f acc = {};
    for (int k = 0; k < FEAT; k += 4) {
        const int ka = k + 2 * kh;
        // A fragment: 16x4 f32, lane holds K=ka, ka+1 (contiguous -> b64 load)
        const float2 af = *(const float2*)(Arow + ka);
        v2f a; a.x = af.x; a.y = af.y;
        // B fragment: 4x16 f32, lane holds rows ka, ka+1 at column n0+mr
        v2f b;
        b.x = B[(size_t)ka * FEAT + n0 + mr];
        b.y = B[(size_t)(ka + 1) * FEAT + n0 + mr];
        acc = __builtin_amdgcn_wmma_f32_16x16x4_f32(
            /*neg_a=*/false, a, /*neg_b=*/false, b,
            /*c_mod=*/(short)0, acc, /*reuse_a=*/false, /*reuse_b=*/false);
    }

    // C/D layout: VGPR v -> row (v + 8*kh), col = mr
    #pragma unroll
    for (int v = 0; v < 8; ++v) {
        C[(size_t)(m0 + v + 8 * kh) * FEAT + n0 + mr] = acc[v];
    }
}

// ---------------------------------------------------------------------------
// SpMM scatter: y[rows[e], :] += vals[e] * x[cols[e], :]
// 256-thread block = 4 edge-groups of 64 threads; each thread handles a
// float4 (64 * 4 = 256 features). Destination must be pre-zeroed.
// Atomics land in L2 (both buffers fit in the 192MB L2).
// ---------------------------------------------------------------------------
__global__ __launch_bounds__(256)
void spmm_scatter(const int*   __restrict__ rows,
                  const int*   __restrict__ cols,
                  const float* __restrict__ vals,
                  const float* __restrict__ xin,
                  float*       __restrict__ yout,
                  int nnz) {
    const int grp  = threadIdx.x >> 6;          // 0..3
    const int lane = threadIdx.x & 63;          // 0..63
    const int e    = blockIdx.x * 4 + grp;
    if (e >= nnz) return;

    const int   r = rows[e];
    const int   c = cols[e];
    const float v = vals[e];

    const float4 t = ((const float4*)(xin + (size_t)c * FEAT))[lane];
    float* yo = yout + (size_t)r * FEAT + lane * 4;
    atomicAdd(yo + 0, v * t.x);
    atomicAdd(yo + 1, v * t.y);
    atomicAdd(yo + 2, v * t.z);
    atomicAdd(yo + 3, v * t.w);
}

// ---------------------------------------------------------------------------
// Soft shrinkage + per-row filter (in place), float4 vectorized.
// ---------------------------------------------------------------------------
__device__ __forceinline__ float softshrink(float x) {
    float m = fabsf(x) - THRESH;
    return m > 0.f ? copysignf(m, x) : 0.f;
}

__global__ __launch_bounds__(256)
void shrink_filter(float* __restrict__ h, const float* __restrict__ filt) {
    const int i = blockIdx.x * blockDim.x + threadIdx.x;   // float4 index
    const int total4 = NNODES * (FEAT / 4);
    if (i >= total4) return;
    const float f = filt[i >> 6];                          // 64 float4 per row
    float4 v = ((float4*)h)[i];
    v.x = softshrink(v.x) * f;
    v.y = softshrink(v.y) * f;
    v.z = softshrink(v.z) * f;
    v.w = softshrink(v.w) * f;
    ((float4*)h)[i] = v;
}

// ---------------------------------------------------------------------------
// Bias add (in place), float4 vectorized.
// ---------------------------------------------------------------------------
__global__ __launch_bounds__(256)
void bias_add(float* __restrict__ h, const float* __restrict__ bias) {
    const int i = blockIdx.x * blockDim.x + threadIdx.x;   // float4 index
    const int total4 = NNODES * (FEAT / 4);
    if (i >= total4) return;
    const float4 b = ((const float4*)bias)[i & 63];
    float4 v = ((float4*)h)[i];
    v.x += b.x; v.y += b.y; v.z += b.z; v.w += b.w;
    ((float4*)h)[i] = v;
}

// ---------------------------------------------------------------------------
// Orchestration. Ping-pong between d_ws (51.2 MB used) and d_out so the
// final pass lands in d_out:
//   gemm -> ws; spmm0 ws->out; spmm1 out->ws; spmm2 ws->out; spmm3 out->ws;
//   shrink(ws); spmm1 ws->out; spmm2 out->ws; spmm3 ws->out; bias(out).
// ---------------------------------------------------------------------------
extern "C" void kernel_launch(void* const* d_in, const int* in_sizes, int n_in,
                              void* d_out, int out_size, void* d_ws, size_t ws_size,
                              hipStream_t stream) {
    const float* x     = (const float*)d_in[0];   // [N, 256]
    const float* W     = (const float*)d_in[1];   // [256, 256]
    const float* filt  = (const float*)d_in[2];   // [200000, 1] (first N used)
    const float* bias  = (const float*)d_in[3];   // [256]
    const float* dvals = (const float*)d_in[4];   // [4, NNZ]
    const int*   drows = (const int*)d_in[5];     // [4, NNZ]
    const int*   dcols = (const int*)d_in[6];     // [4, NNZ]

    float* out = (float*)d_out;
    float* ws  = (float*)d_ws;                    // needs N*FEAT*4 = 51.2 MB
    (void)in_sizes; (void)n_in; (void)out_size; (void)ws_size;

    const size_t hbytes   = (size_t)NNODES * FEAT * sizeof(float);
    const int    ew_total = NNODES * (FEAT / 4);
    const dim3   ew_grid((ew_total + 255) / 256);
    const dim3   sp_grid((NNZ_OP + 3) / 4);

    // 1. h = x @ W  -> ws   (fp32 WMMA)
    gemm_f32_wmma<<<dim3(NNODES / 16, 2), 256, 0, stream>>>(x, W, ws);

    auto spmm = [&](int op, const float* src, float* dst) {
        hipMemsetAsync(dst, 0, hbytes, stream);
        spmm_scatter<<<sp_grid, 256, 0, stream>>>(
            drows + (size_t)op * NNZ_OP,
            dcols + (size_t)op * NNZ_OP,
            dvals + (size_t)op * NNZ_OP,
            src, dst, NNZ_OP);
    };

    // 2. decomposition: ops 0..3
    spmm(0, ws, out);
    spmm(1, out, ws);
    spmm(2, ws, out);
    spmm(3, out, ws);

    // 3. soft shrink * filter (in ws)
    shrink_filter<<<ew_grid, 256, 0, stream>>>(ws, filt);

    // 4. reconstruction: ops 1..3
    spmm(1, ws, out);
    spmm(2, out, ws);
    spmm(3, ws, out);

    // 5. + bias (in d_out)
    bias_add<<<ew_grid, 256, 0, stream>>>(out, bias);
}